// DirSAGEEmbRes_1151051235613
// MI455X (gfx1250) — compile-verified
//
#include <hip/hip_runtime.h>

typedef float v2f __attribute__((ext_vector_type(2)));
typedef float v8f __attribute__((ext_vector_type(8)));

#define HID 128

// ---------------------------------------------------------------------------
// Degree counts: cnt[dst] += 1.0f for both edge sets (matches segment_sum of 1s)
// ---------------------------------------------------------------------------
__global__ void k_degrees(const int* __restrict__ ein, const int* __restrict__ eout,
                          int E, float* cin, float* cout_) {
  int i = blockIdx.x * blockDim.x + threadIdx.x;
  if (i < E) {
    atomicAdd(&cin[ein[E + i]], 1.0f);
  } else if (i < 2 * E) {
    int j = i - E;
    atomicAdd(&cout_[eout[E + j]], 1.0f);
  }
}

// ---------------------------------------------------------------------------
// Pack W2 = [Wl(K0 x 128); Wr(Ktot-K0 x 128)] into WMMA f32 16x16x4 B-fragment
// layout: for k-step s, n-tile t, lane l holds B[s*4 + 2*(l>>4) + j][t*16+(l&15)]
// at dst[((s*8+t)*32 + l)*2 + j]  -> GEMM reads one coalesced b64 per lane.
// ---------------------------------------------------------------------------
__global__ void k_pack_w(const float* __restrict__ Wl, const float* __restrict__ Wr,
                         int K0, float* __restrict__ dst, int total) {
  int idx = blockIdx.x * blockDim.x + threadIdx.x;
  if (idx >= total) return;
  int lane = idx & 31;
  int t = (idx >> 5) & 7;
  int s = idx >> 8;
  int n = t * 16 + (lane & 15);
  int kb = s * 4 + ((lane >> 4) << 1);
  float v0 = (kb     < K0) ? Wl[kb * HID + n]           : Wr[(kb - K0) * HID + n];
  float v1 = (kb + 1 < K0) ? Wl[(kb + 1) * HID + n]     : Wr[(kb + 1 - K0) * HID + n];
  dst[idx * 2]     = v0;
  dst[idx * 2 + 1] = v1;
}

// ---------------------------------------------------------------------------
// Edge scatter-add: sums[dst] += feat[src], feat is virtual concat of p0|p1
// (float4 chunks). Consecutive threads walk one source row -> coalesced.
// ---------------------------------------------------------------------------
__global__ void k_scatter(const int* __restrict__ edge, int E,
                          const float4* __restrict__ p0, int c0,
                          const float4* __restrict__ p1, int c1,
                          float* __restrict__ sums, int dim) {
  int idx = blockIdx.x * blockDim.x + threadIdx.x;
  int C = c0 + c1;
  if (idx >= E * C) return;
  int e = idx / C;
  int c = idx - e * C;
  int s = edge[e];        // src row of edge e
  int d = edge[E + e];    // dst row of edge e
  float4 v = (c < c0) ? p0[(size_t)s * c0 + c] : p1[(size_t)s * c1 + (c - c0)];
  float* o = sums + (size_t)d * dim + c * 4;
  atomicAdd(o + 0, v.x);
  atomicAdd(o + 1, v.y);
  atomicAdd(o + 2, v.z);
  atomicAdd(o + 3, v.w);
}

// ---------------------------------------------------------------------------
// WMMA GEMM: out[N,128] = relu( A[N,Ktot] @ W2[Ktot,128] + bias ) (+ residual)
// A is virtual: k<K0        -> msum[row][k] * (1/max(cnt[row],1))   (mean)
//               k<K0+K1     -> src1[row][k-K0]
//               else        -> src2[row][k-K0-K1]
// Region boundaries are multiples of 4 and khalf<=2, so the region per k-step
// is wave-uniform: split into 3 uniform loops (no divergent exec masking).
// One wave computes a 16x128 tile with 8 x v_wmma_f32_16x16x4_f32 per k-step.
// addRes: out = relu(acc+bias) + src1_row (layer-2 residual, in-place safe:
// each wave only reads/writes its own 16 rows).
// ---------------------------------------------------------------------------
__global__ __launch_bounds__(256) void k_gemm(
    const float* __restrict__ msum, const float* __restrict__ cnt,
    const float* src1, const float* __restrict__ src2,
    const float* __restrict__ Wp, const float* __restrict__ bias,
    float* out, int N, int K0, int K1, int K2e, int addRes)
{
  int wave = blockIdx.x * (blockDim.x >> 5) + (threadIdx.x >> 5);
  int lane = threadIdx.x & 31;
  int r0 = wave * 16;
  if (r0 >= N) return;

  int row   = r0 + (lane & 15);         // A-fragment row for this lane
  int khalf = (lane >> 4) << 1;         // lanes 16-31 hold K+2,K+3
  float inv = 1.0f / fmaxf(cnt[row], 1.0f);

  v8f acc[8] = {};

  auto mma8 = [&](v2f a, const float* wk) {
#pragma unroll
    for (int t = 0; t < 8; ++t) {
      v2f b = *(const v2f*)(wk + t * 64);
      acc[t] = __builtin_amdgcn_wmma_f32_16x16x4_f32(
          false, a, false, b, (short)0, acc[t], false, false);
    }
  };

  const float* wk  = Wp + lane * 2;                     // advances 512 per k-step
  const float* pA0 = msum + (size_t)row * K0 + khalf;   // mean-sum region
  for (int k = 0; k < K0; k += 4, wk += 512) {
    v2f a = *(const v2f*)(pA0 + k);
    a[0] *= inv;
    a[1] *= inv;
    mma8(a, wk);
  }
  const float* pA1 = src1 + (size_t)row * K1 + khalf;   // self-feature region
  for (int k = 0; k < K1; k += 4, wk += 512) {
    v2f a = *(const v2f*)(pA1 + k);
    mma8(a, wk);
  }
  if (K2e > 0) {
    const float* pA2 = src2 + (size_t)row * K2e + khalf; // emb region (layer 1)
    for (int k = 0; k < K2e; k += 4, wk += 512) {
      v2f a = *(const v2f*)(pA2 + k);
      mma8(a, wk);
    }
  }

  // Epilogue: C/D layout -> lane l, vgpr v: M = r0 + (l>>4)*8 + v, N = t*16+(l&15)
  int n0 = lane & 15;
  int mA = (lane >> 4) << 3;
#pragma unroll
  for (int t = 0; t < 8; ++t) {
    float bv = bias[t * 16 + n0];
#pragma unroll
    for (int v = 0; v < 8; ++v) {
      int m = r0 + mA + v;
      int n = t * 16 + n0;
      float val = fmaxf(acc[t][v] + bv, 0.0f);
      if (addRes) val += src1[(size_t)m * HID + n];
      out[(size_t)m * HID + n] = val;
    }
  }
}

// ---------------------------------------------------------------------------
// Final linear: y[i] = [h_in|h_out] . linW + b ; one wave per node.
// ---------------------------------------------------------------------------
__global__ void k_final(const float* __restrict__ hin, const float* __restrict__ hout,
                        const float* __restrict__ W, const float* __restrict__ b,
                        float* __restrict__ y, int N) {
  int node = blockIdx.x * (blockDim.x >> 5) + (threadIdx.x >> 5);
  int lane = threadIdx.x & 31;
  if (node >= N) return;
  const float* hi = hin  + (size_t)node * HID;
  const float* ho = hout + (size_t)node * HID;
  float s = 0.f;
#pragma unroll
  for (int c = 0; c < HID; c += 32) {
    s += hi[c + lane] * W[c + lane];
    s += ho[c + lane] * W[HID + c + lane];
  }
#pragma unroll
  for (int off = 16; off; off >>= 1) s += __shfl_xor(s, off, 32);
  if (lane == 0) y[node] = s + b[0];
}

// ---------------------------------------------------------------------------
extern "C" void kernel_launch(void* const* d_in, const int* in_sizes, int n_in,
                              void* d_out, int out_size, void* d_ws, size_t ws_size,
                              hipStream_t stream) {
  const float* x       = (const float*)d_in[0];
  const int*   ein     = (const int*)d_in[1];
  const int*   eout    = (const int*)d_in[2];
  const float* emb     = (const float*)d_in[3];
  const float* in1_Wl  = (const float*)d_in[4];
  const float* in1_bl  = (const float*)d_in[5];
  const float* in1_Wr  = (const float*)d_in[6];
  const float* in2_Wl  = (const float*)d_in[7];
  const float* in2_bl  = (const float*)d_in[8];
  const float* in2_Wr  = (const float*)d_in[9];
  const float* out1_Wl = (const float*)d_in[10];
  const float* out1_bl = (const float*)d_in[11];
  const float* out1_Wr = (const float*)d_in[12];
  const float* out2_Wl = (const float*)d_in[13];
  const float* out2_bl = (const float*)d_in[14];
  const float* out2_Wr = (const float*)d_in[15];
  const float* linW    = (const float*)d_in[16];
  const float* linb    = (const float*)d_in[17];

  const int N  = in_sizes[0] / 128;   // IN_DIM = 128
  const int E  = in_sizes[1] / 2;     // edge = [2, E]
  const int D0 = 144;                 // IN_DIM + EMB_DIM

  // ---- workspace carve-out (256B aligned slots) ----
  char* ws = (char*)d_ws;
  size_t off = 0;
  auto carve = [&](size_t bytes) -> void* {
    void* p = ws + off;
    off = (off + bytes + 255) & ~(size_t)255;
    return p;
  };
  float* cnt_in  = (float*)carve((size_t)N * 4);
  float* cnt_out = (float*)carve((size_t)N * 4);
  float* msum    = (float*)carve((size_t)N * D0 * 4);   // reused by all 4 aggs
  float* hin     = (float*)carve((size_t)N * HID * 4);  // h_in1 -> h_in (in place)
  float* hout    = (float*)carve((size_t)N * HID * 4);  // h_out1 -> h_out (in place)
  float* Wp1     = (float*)carve((size_t)(288 / 4) * 512 * 4);
  float* Wp2     = (float*)carve((size_t)(256 / 4) * 512 * 4);
  float* Wp3     = (float*)carve((size_t)(288 / 4) * 512 * 4);
  float* Wp4     = (float*)carve((size_t)(256 / 4) * 512 * 4);
  (void)ws_size; (void)n_in; (void)out_size;

  // ---- degrees ----
  hipMemsetAsync(cnt_in, 0, (size_t)N * 4, stream);
  hipMemsetAsync(cnt_out, 0, (size_t)N * 4, stream);
  k_degrees<<<(2 * E + 255) / 256, 256, 0, stream>>>(ein, eout, E, cnt_in, cnt_out);

  // ---- weight repack into WMMA B layout ----
  k_pack_w<<<72, 256, 0, stream>>>(in1_Wl,  in1_Wr,  144, Wp1, 72 * 256);
  k_pack_w<<<64, 256, 0, stream>>>(in2_Wl,  in2_Wr,  128, Wp2, 64 * 256);
  k_pack_w<<<72, 256, 0, stream>>>(out1_Wl, out1_Wr, 144, Wp3, 72 * 256);
  k_pack_w<<<64, 256, 0, stream>>>(out2_Wl, out2_Wr, 128, Wp4, 64 * 256);

  const int waves   = (N + 15) / 16;
  const int gblocks = (waves + 7) / 8;   // 8 waves (256 threads) per block

  // ---- IN direction, layer 1 ----
  hipMemsetAsync(msum, 0, (size_t)N * 144 * 4, stream);
  k_scatter<<<(E * 36 + 255) / 256, 256, 0, stream>>>(
      ein, E, (const float4*)x, 32, (const float4*)emb, 4, msum, 144);
  k_gemm<<<gblocks, 256, 0, stream>>>(msum, cnt_in, x, emb, Wp1, in1_bl,
                                      hin, N, 144, 128, 16, 0);
  // ---- IN direction, layer 2 (+residual, in place) ----
  hipMemsetAsync(msum, 0, (size_t)N * 128 * 4, stream);
  k_scatter<<<(E * 32 + 255) / 256, 256, 0, stream>>>(
      ein, E, (const float4*)hin, 32, (const float4*)hin, 0, msum, 128);
  k_gemm<<<gblocks, 256, 0, stream>>>(msum, cnt_in, hin, nullptr, Wp2, in2_bl,
                                      hin, N, 128, 128, 0, 1);

  // ---- OUT direction, layer 1 ----
  hipMemsetAsync(msum, 0, (size_t)N * 144 * 4, stream);
  k_scatter<<<(E * 36 + 255) / 256, 256, 0, stream>>>(
      eout, E, (const float4*)x, 32, (const float4*)emb, 4, msum, 144);
  k_gemm<<<gblocks, 256, 0, stream>>>(msum, cnt_out, x, emb, Wp3, out1_bl,
                                      hout, N, 144, 128, 16, 0);
  // ---- OUT direction, layer 2 (+residual, in place) ----
  hipMemsetAsync(msum, 0, (size_t)N * 128 * 4, stream);
  k_scatter<<<(E * 32 + 255) / 256, 256, 0, stream>>>(
      eout, E, (const float4*)hout, 32, (const float4*)hout, 0, msum, 128);
  k_gemm<<<gblocks, 256, 0, stream>>>(msum, cnt_out, hout, nullptr, Wp4, out2_bl,
                                      hout, N, 128, 128, 0, 1);

  // ---- final linear ----
  k_final<<<(N + 7) / 8, 256, 0, stream>>>(hin, hout, linW, linb, (float*)d_out, N);
}